// CenterLoss_67783173866135
// MI455X (gfx1250) — compile-verified
//
#include <hip/hip_runtime.h>

// ---------------------------------------------------------------------------
// CenterLoss on MI455X (gfx1250, wave32).
//
// Reference algebraically reduces to:
//   loss = mean_b clip(1 - dot(f_b, c_tb)/(|f_b||c_tb|), EPS, MAXV) + (C-1)*EPS
// (the one-hot mask keeps one distance per row; every masked-off zero is
//  clipped up to EPS by jnp.clip).
//
// Kernel 1: one wave (32 lanes) per 16-row tile.
//   - 16 dot products on the matrix pipe with V_WMMA_F32_16X16X4_F32,
//     K=512 accumulated in 128 WMMA issues; diagonal of the 16x16 result
//     gives the wanted dots in full f32 precision.
//   - row norms computed from the same fragments (lane (half,r) sees exactly
//     K = {2h,2h+1} mod 4 of row r) -> every element loaded exactly once.
//   - explicit ping-pong software pipeline: 2 register buffers of 8
//     fragment-pairs; a 16-load burst fills one buffer while the other is
//     consumed by WMMA + norm FMAs, keeping ~16 loads in flight.
// Kernel 2: deterministic tree-reduce of the 256 partials + constant term.
// ---------------------------------------------------------------------------

typedef float v2f __attribute__((ext_vector_type(2)));
typedef float v8f __attribute__((ext_vector_type(8)));

#define NUM_CLASSES 10000
#define EMBED       512
#define BATCH       4096
#define EPS_F       1e-12f
#define MAXV_F      1e12f
#define TILE        16
#define NTILES      (BATCH / TILE)   // 256
#define PF          8                // fragment-pairs per pipeline buffer
#define NBLK        (EMBED / (4 * PF))  // 16 K-blocks of 32 K-values

__global__ __launch_bounds__(32)
void center_loss_tile_kernel(const float* __restrict__ F,   // [BATCH][EMBED]
                             const float* __restrict__ C,   // [NUM_CLASSES][EMBED]
                             const int*   __restrict__ T,   // [BATCH]
                             float*       __restrict__ partial) // [NTILES]
{
    const int tile = blockIdx.x;
    const int lane = threadIdx.x;     // 0..31, full wave active everywhere
    const int half = lane >> 4;       // K sub-group: lanes 16-31 carry K=2,3
    const int r    = lane & 15;       // A: M-row index, B: N-col index in tile
    const int row  = tile * TILE + r;

    const float* __restrict__ frow = F + (size_t)row * EMBED;
    const int    tgt               = T[row];
    const float* __restrict__ crow = C + (size_t)tgt * EMBED;
    const int    kb                = 2 * half;

    // fragment-group g covers K = 4g .. 4g+3; this lane owns {4g+kb, 4g+kb+1}
#define LDA(g) (*(const v2f*)(frow + 4 * (g) + kb))
#define LDB(g) (*(const v2f*)(crow + 4 * (g) + kb))

    v8f   acc = {};
    float nf  = 0.0f, nc = 0.0f;

#define CONSUME(ai, bi)                                                       \
    do {                                                                      \
        v2f _a = (ai), _b = (bi);                                             \
        nf += _a.x * _a.x + _a.y * _a.y;                                      \
        nc += _b.x * _b.x + _b.y * _b.y;                                      \
        acc = __builtin_amdgcn_wmma_f32_16x16x4_f32(                          \
                  false, _a, false, _b, (short)0, acc, false, false);         \
    } while (0)

    v2f A0[PF], B0[PF], A1[PF], B1[PF];

#pragma unroll
    for (int i = 0; i < PF; ++i) { A0[i] = LDA(i); B0[i] = LDB(i); }

#pragma unroll 1
    for (int blk = 0; blk < NBLK - 2; blk += 2) {
#pragma unroll
        for (int i = 0; i < PF; ++i) {
            A1[i] = LDA((blk + 1) * PF + i);
            B1[i] = LDB((blk + 1) * PF + i);
        }
#pragma unroll
        for (int i = 0; i < PF; ++i) CONSUME(A0[i], B0[i]);
#pragma unroll
        for (int i = 0; i < PF; ++i) {
            A0[i] = LDA((blk + 2) * PF + i);
            B0[i] = LDB((blk + 2) * PF + i);
        }
#pragma unroll
        for (int i = 0; i < PF; ++i) CONSUME(A1[i], B1[i]);
    }
    // epilogue: buffer0 holds block NBLK-2; fetch last block, drain both
#pragma unroll
    for (int i = 0; i < PF; ++i) {
        A1[i] = LDA((NBLK - 1) * PF + i);
        B1[i] = LDB((NBLK - 1) * PF + i);
    }
#pragma unroll
    for (int i = 0; i < PF; ++i) CONSUME(A0[i], B0[i]);
#pragma unroll
    for (int i = 0; i < PF; ++i) CONSUME(A1[i], B1[i]);

#undef LDA
#undef LDB
#undef CONSUME

    // complete the norms across the two K half-groups
    nf += __shfl_xor(nf, 16, 32);
    nc += __shfl_xor(nc, 16, 32);

    // ---- diagonal extract: lane (half, r) holds M = v + 8*half, N = r ----
    float contrib = 0.0f;
    const int v = r - 8 * half;
    if (v >= 0 && v < 8) {
        float dot = 0.0f;
#pragma unroll
        for (int i = 0; i < 8; ++i)
            if (v == i) dot = acc[i];
        float d = 1.0f - dot * __frsqrt_rn(nf) * __frsqrt_rn(nc);
        d = fminf(fmaxf(d, EPS_F), MAXV_F);
        contrib = d;
    }

    // ---- wave-level sum of the 16 diagonal contributions ----
#pragma unroll
    for (int off = 16; off > 0; off >>= 1)
        contrib += __shfl_xor(contrib, off, 32);

    if (lane == 0)
        partial[tile] = contrib;
}

__global__ __launch_bounds__(NTILES)
void center_loss_reduce_kernel(const float* __restrict__ partial,
                               float*       __restrict__ out)
{
    __shared__ float s[NTILES];
    s[threadIdx.x] = partial[threadIdx.x];
    __syncthreads();
#pragma unroll
    for (int off = NTILES / 2; off > 0; off >>= 1) {
        if ((int)threadIdx.x < off)
            s[threadIdx.x] += s[threadIdx.x + off];
        __syncthreads();
    }
    if (threadIdx.x == 0) {
        // every masked-off entry was clipped up to EPS:
        //   + B*(C-1)*EPS summed, then / B  ->  + (C-1)*EPS
        out[0] = s[0] / (float)BATCH + (float)(NUM_CLASSES - 1) * EPS_F;
    }
}

extern "C" void kernel_launch(void* const* d_in, const int* in_sizes, int n_in,
                              void* d_out, int out_size, void* d_ws, size_t ws_size,
                              hipStream_t stream)
{
    const float* F = (const float*)d_in[0];   // features [4096,512] f32
    const float* C = (const float*)d_in[1];   // centers  [10000,512] f32
    const int*   T = (const int*)d_in[2];     // targets  [4096] i32
    float*       out     = (float*)d_out;     // scalar loss
    float*       partial = (float*)d_ws;      // NTILES floats of scratch

    center_loss_tile_kernel<<<NTILES, 32, 0, stream>>>(F, C, T, partial);
    center_loss_reduce_kernel<<<1, NTILES, 0, stream>>>(partial, out);
}